// ComputeMetrics_15444702397236
// MI455X (gfx1250) — compile-verified
//
#include <hip/hip_runtime.h>
#include <stdint.h>

// ---------------------------------------------------------------------------
// Problem constants (reference: logits [2,2048,32000] f32, labels [2,2048])
// ---------------------------------------------------------------------------
#define VOCAB   32000
#define S_LEN   2048
#define BTH     512                     // threads per row-block (16 waves of 32)
#define CHUNK   (BTH * 4)               // 2048 floats staged per chunk (8 KB)
#define NFULL   (VOCAB / CHUNK)         // 15 full chunks (30720 floats)
#define TAIL    (VOCAB - NFULL * CHUNK) // 1280 float tail, direct loads
#define DEPTH   4                       // async ring depth: 3 outstanding/wave
#define LOG2E   1.4426950408889634f

#if defined(__has_builtin)
#if __has_builtin(__builtin_amdgcn_exp2f)
#define EXP2F(x) __builtin_amdgcn_exp2f(x)
#endif
#endif
#ifndef EXP2F
#define EXP2F(x) exp2f(x)
#endif

// ---------------------------------------------------------------------------
// CDNA5 async global -> LDS copy (16 B per lane), tracked by ASYNCcnt.
// The addrspace(3) offset is the low 32 bits of the flat pointer.
// ---------------------------------------------------------------------------
__device__ __forceinline__ void async_copy_b128(const float* gsrc, float* ldst) {
    uint32_t lds_addr = (uint32_t)(uintptr_t)ldst;   // wave-relative LDS byte addr
    uint64_t gaddr    = (uint64_t)(uintptr_t)gsrc;
    asm volatile("global_load_async_to_lds_b128 %0, %1, off"
                 :: "v"(lds_addr), "v"(gaddr)
                 : "memory");
}

__device__ __forceinline__ void wait_async_le3() { asm volatile("s_wait_asynccnt 3" ::: "memory"); }
__device__ __forceinline__ void wait_async_le2() { asm volatile("s_wait_asynccnt 2" ::: "memory"); }
__device__ __forceinline__ void wait_async_le1() { asm volatile("s_wait_asynccnt 1" ::: "memory"); }
__device__ __forceinline__ void wait_async_le0() { asm volatile("s_wait_asynccnt 0" ::: "memory"); }

// ---------------------------------------------------------------------------
// Kernel 1: one workgroup per token row.
// Streams the 125 KB row through a 4-deep async-DMA LDS ring (3 outstanding
// 512 B ops per wave for MLP), computes online (max, Z, Σ(x-m)e^{x-m}) +
// rank counts, reduces across the block, writes per-row ce / entropy into
// d_out and packed hit flags into d_ws.
// ---------------------------------------------------------------------------
__global__ __launch_bounds__(BTH) void CeRow(const float* __restrict__ logits,
                                             const int*   __restrict__ labels,
                                             float*       __restrict__ out,
                                             uint32_t*    __restrict__ flags,
                                             int nrows) {
    __shared__ float    smem[DEPTH][CHUNK];
    __shared__ float    red_m[BTH];
    __shared__ float    red_s[BTH];
    __shared__ float    red_u[BTH];
    __shared__ uint32_t red_c[BTH];

    const int row = blockIdx.x;
    const int tid = threadIdx.x;
    const float* __restrict__ rowp = logits + (size_t)row * VOCAB;

    // Shifted label: labels[b, s+1], IGNORE(-100) at s == S-1.
    const int sp  = row & (S_LEN - 1);
    const int lab = (sp < S_LEN - 1) ? labels[row + 1] : -100;

    if (lab < 0) {  // invalid token: zero outputs, excluded from means
        if (tid == 0) {
            out[1 + row]         = 0.0f;
            out[1 + nrows + row] = 0.0f;
            flags[row]           = 0u;
        }
        return;
    }

    const float xl = rowp[lab];   // uniform address -> broadcast load

    // Hoist the tail's direct global load so it overlaps the whole pipeline.
    float4 xt = make_float4(0.f, 0.f, 0.f, 0.f);
    const bool has_tail = (tid * 4) < TAIL;
    if (has_tail)
        xt = *reinterpret_cast<const float4*>(rowp + NFULL * CHUNK + tid * 4);

    float m = -1.0e30f, s = 0.0f, u = 0.0f;   // online softmax state
    uint32_t cgt = 0, ceq = 0;                // rank counters

    auto consume = [&](const float4& x, int col) {
        // Rescale only when chunk max exceeds running max (rare after warm-up).
        const float mc = fmaxf(fmaxf(x.x, x.y), fmaxf(x.z, x.w));
        if (mc > m) {
            const float cc = EXP2F((m - mc) * LOG2E);
            u = (u + (m - mc) * s) * cc;
            s = s * cc;
            m = mc;
        }
        const float d0 = x.x - m, d1 = x.y - m, d2 = x.z - m, d3 = x.w - m;
        const float e0 = EXP2F(d0 * LOG2E), e1 = EXP2F(d1 * LOG2E);
        const float e2 = EXP2F(d2 * LOG2E), e3 = EXP2F(d3 * LOG2E);
        s += (e0 + e1) + (e2 + e3);
        u += (d0 * e0 + d1 * e1) + (d2 * e2 + d3 * e3);
        cgt += (x.x > xl) + (x.y > xl) + (x.z > xl) + (x.w > xl);
        ceq += ((x.x == xl) & (col + 0 < lab)) + ((x.y == xl) & (col + 1 < lab)) +
               ((x.z == xl) & (col + 2 < lab)) + ((x.w == xl) & (col + 3 < lab));
    };

    // Prologue: fill the ring with chunks 0..2 (3 outstanding asyncs).
    async_copy_b128(rowp + 0 * CHUNK + tid * 4, &smem[0][tid * 4]);
    async_copy_b128(rowp + 1 * CHUNK + tid * 4, &smem[1][tid * 4]);
    async_copy_b128(rowp + 2 * CHUNK + tid * 4, &smem[2][tid * 4]);

    // Steady state: issue chunk c+3, wait <=3 (=> chunk c complete, in-order).
    for (int c = 0; c < NFULL - (DEPTH - 1); ++c) {
        async_copy_b128(rowp + (size_t)(c + DEPTH - 1) * CHUNK + tid * 4,
                        &smem[(c + DEPTH - 1) & (DEPTH - 1)][tid * 4]);
        wait_async_le3();
        const float4 x = *reinterpret_cast<const float4*>(&smem[c & (DEPTH - 1)][tid * 4]);
        consume(x, c * CHUNK + tid * 4);
    }
    // Drain the ring: chunks NFULL-3, NFULL-2, NFULL-1.
    {
        wait_async_le2();
        const int c = NFULL - 3;
        consume(*reinterpret_cast<const float4*>(&smem[c & (DEPTH - 1)][tid * 4]),
                c * CHUNK + tid * 4);
    }
    {
        wait_async_le1();
        const int c = NFULL - 2;
        consume(*reinterpret_cast<const float4*>(&smem[c & (DEPTH - 1)][tid * 4]),
                c * CHUNK + tid * 4);
    }
    {
        wait_async_le0();
        const int c = NFULL - 1;
        consume(*reinterpret_cast<const float4*>(&smem[c & (DEPTH - 1)][tid * 4]),
                c * CHUNK + tid * 4);
    }

    // Tail (1280 floats), loaded long ago into registers.
    if (has_tail)
        consume(xt, NFULL * CHUNK + tid * 4);

    // Block-wide combine of (m, s, u) triples and counts.
    red_m[tid] = m;
    red_s[tid] = s;
    red_u[tid] = u;
    red_c[tid] = (cgt & 0xFFFFu) | (ceq << 16);   // each count < 32000 -> 16 bits
    __syncthreads();

    for (int off = BTH / 2; off > 0; off >>= 1) {
        if (tid < off) {
            const float ma = red_m[tid],       mb = red_m[tid + off];
            const float sa = red_s[tid],       sb = red_s[tid + off];
            const float ua = red_u[tid],       ub = red_u[tid + off];
            const float M  = fmaxf(ma, mb);
            const float ca = EXP2F((ma - M) * LOG2E);
            const float cb = EXP2F((mb - M) * LOG2E);
            red_m[tid] = M;
            red_s[tid] = sa * ca + sb * cb;
            red_u[tid] = (ua + (ma - M) * sa) * ca + (ub + (mb - M) * sb) * cb;
            red_c[tid] = red_c[tid] + red_c[tid + off];
        }
        __syncthreads();
    }

    if (tid == 0) {
        const float M = red_m[0], Z = red_s[0], U = red_u[0];
        const float lnZ = logf(Z);
        out[1 + row]         = M + lnZ - xl;      // ce = -(x_l - M - ln Z)
        out[1 + nrows + row] = lnZ - U / Z;       // entropy = ln Z - E[x-M]
        const uint32_t c    = red_c[0];
        const uint32_t rank = (c & 0xFFFFu) + (c >> 16);
        uint32_t f = 1u;                          // bit0: valid
        if (rank < 1u)  f |= 2u;                  // top-1
        if (rank < 5u)  f |= 4u;                  // top-5
        if (rank < 20u) f |= 8u;                  // top-20
        flags[row] = f;
    }
}

// ---------------------------------------------------------------------------
// Kernel 2: single-block deterministic finalize (no atomics, graph-safe).
// ---------------------------------------------------------------------------
__global__ __launch_bounds__(1024) void Finalize(float* __restrict__ out,
                                                 const uint32_t* __restrict__ flags,
                                                 int nrows) {
    __shared__ float              s_ce[1024];
    __shared__ unsigned long long s_cnt[1024];

    const int tid = threadIdx.x;
    float ce = 0.0f;
    unsigned long long cnt = 0ull;   // 16-bit fields: valid | h1<<16 | h5<<32 | h20<<48
    for (int r = tid; r < nrows; r += 1024) {
        ce += out[1 + r];
        const uint32_t f = flags[r];
        cnt += (unsigned long long)(f & 1u)
             | ((unsigned long long)((f >> 1) & 1u) << 16)
             | ((unsigned long long)((f >> 2) & 1u) << 32)
             | ((unsigned long long)((f >> 3) & 1u) << 48);
    }
    s_ce[tid]  = ce;
    s_cnt[tid] = cnt;
    __syncthreads();
    for (int off = 512; off > 0; off >>= 1) {
        if (tid < off) {
            s_ce[tid]  += s_ce[tid + off];
            s_cnt[tid] += s_cnt[tid + off];
        }
        __syncthreads();
    }
    if (tid == 0) {
        const unsigned long long c = s_cnt[0];
        const float nv = (float)(c & 0xFFFFull);
        out[0]                 = s_ce[0] / nv;
        out[1 + 2 * nrows + 0] = (float)((c >> 16) & 0xFFFFull) / nv;
        out[1 + 2 * nrows + 1] = (float)((c >> 32) & 0xFFFFull) / nv;
        out[1 + 2 * nrows + 2] = (float)((c >> 48) & 0xFFFFull) / nv;
    }
}

// ---------------------------------------------------------------------------
extern "C" void kernel_launch(void* const* d_in, const int* in_sizes, int n_in,
                              void* d_out, int out_size, void* d_ws, size_t ws_size,
                              hipStream_t stream) {
    const float* logits = (const float*)d_in[0];
    const int*   labels = (const int*)d_in[1];
    float*       out    = (float*)d_out;
    uint32_t*    flags  = (uint32_t*)d_ws;       // nrows * 4 bytes of scratch

    const int nrows = in_sizes[1];               // B*S = 4096

    CeRow<<<nrows, BTH, 0, stream>>>(logits, labels, out, flags, nrows);
    Finalize<<<1, 1024, 0, stream>>>(out, flags, nrows);
}